// BinaryAssociativeMemory_15702400434864
// MI455X (gfx1250) — compile-verified
//
#include <hip/hip_runtime.h>

// ---------------------------------------------------------------------------
// BinaryAssociativeMemory on MI455X (gfx1250), fp32 end-to-end via
// V_WMMA_F32_16X16X4_F32.  ~146 GFLOP fp32, dominated by the two dense GEMMs
// (103G qkv + 34G W_o).  Strip GEMMs are software-pipelined with a parity-
// indexed ping-pong register buffer (unroll 2) so the compiler emits no
// buffer-rotation moves: loads land directly in the registers the next
// compute stage consumes, with staggered s_wait_loadcnt.
// ---------------------------------------------------------------------------

#define B_   4
#define T_   4096
#define D_   1024
#define H_   16
#define DH_  64
#define C_   64
#define NC_  64   // T_/C_

typedef float v2f __attribute__((ext_vector_type(2)));
typedef float v8f __attribute__((ext_vector_type(8)));

__device__ __forceinline__ v8f wmma4(v2f a, v2f b, v8f c) {
  // D(16x16,f32) = A(16x4,f32) * B(4x16,f32) + C
  return __builtin_amdgcn_wmma_f32_16x16x4_f32(
      /*neg_a=*/false, a, /*neg_b=*/false, b,
      /*c_mod=*/(short)0, c, /*reuse_a=*/false, /*reuse_b=*/false);
}

__device__ __forceinline__ void load_block(const float* __restrict__ arow,
                                           const float* __restrict__ w0,
                                           const float* __restrict__ w1,
                                           const float* __restrict__ w2,
                                           const float* __restrict__ w3,
                                           const int base,  // kb + 2*(lane/16)
                                           v2f a[4], v2f b[4][4]) {
#pragma unroll
  for (int s = 0; s < 4; ++s) {
    const int ko = base + 4 * s;
    a[s]    = *(const v2f*)(arow + ko);
    b[s][0] = *(const v2f*)(w0 + ko);
    b[s][1] = *(const v2f*)(w1 + ko);
    b[s][2] = *(const v2f*)(w2 + ko);
    b[s][3] = *(const v2f*)(w3 + ko);
  }
}

// ---------------------------------------------------------------------------
// Ping-pong double-buffered 16(M) x 64(N) strip GEMM core.  One wave.
// K-block = 16 K-values (4 WMMA steps x 4 N-tiles = 16 WMMAs per block).
// ---------------------------------------------------------------------------
__device__ __forceinline__ void strip_gemm(const float* __restrict__ arow,
                                           const float* __restrict__ w0,
                                           const float* __restrict__ w1,
                                           const float* __restrict__ w2,
                                           const float* __restrict__ w3,
                                           const int ko0,   // 2*(lane/16)
                                           v8f acc[4]) {
  v2f aB[2][4], bB[2][4][4];
  load_block(arow, w0, w1, w2, w3, ko0, aB[0], bB[0]);

#pragma unroll 2
  for (int kb = 0; kb < D_; kb += 16) {
    const int cur = (kb >> 4) & 1;   // compile-time after unroll-2
    const int nx  = cur ^ 1;
    const int nxt = kb + 16;
    if (nxt < D_)
      load_block(arow, w0, w1, w2, w3, nxt + ko0, aB[nx], bB[nx]);
#pragma unroll
    for (int s = 0; s < 4; ++s) {
      acc[0] = wmma4(aB[cur][s], bB[cur][s][0], acc[0]);
      acc[1] = wmma4(aB[cur][s], bB[cur][s][1], acc[1]);
      acc[2] = wmma4(aB[cur][s], bB[cur][s][2], acc[2]);
      acc[3] = wmma4(aB[cur][s], bB[cur][s][3], acc[3]);
    }
  }
}

// ---------------------------------------------------------------------------
// Kernel 1: qkv = x @ W_qkv^T, fused scale(q) + binarize(k,v), scatter into
// head-major [B,H,T,64] workspaces.  4 waves/block, each wave: 16(M) x 64(N).
// ---------------------------------------------------------------------------
__global__ __launch_bounds__(128)
void qkv_gemm_kernel(const float* __restrict__ x,
                     const float* __restrict__ Wqkv,
                     float* __restrict__ q_ws,
                     float* __restrict__ k_ws,
                     float* __restrict__ v_ws) {
  const int lane = threadIdx.x & 31;
  const int wave = threadIdx.x >> 5;
  const int ml   = lane & 15;
  const int kh   = lane >> 4;

  const int m0 = (blockIdx.y * 4 + wave) * 16;
  const int n0 = blockIdx.x * 64;

  v8f acc[4] = {{}, {}, {}, {}};
  strip_gemm(x    + (size_t)(m0 + ml) * D_,
             Wqkv + (size_t)(n0 +  0 + ml) * D_,
             Wqkv + (size_t)(n0 + 16 + ml) * D_,
             Wqkv + (size_t)(n0 + 32 + ml) * D_,
             Wqkv + (size_t)(n0 + 48 + ml) * D_,
             2 * kh, acc);

  const float scale = 0.125f;  // 1/sqrt(64)
#pragma unroll
  for (int t = 0; t < 4; ++t) {
#pragma unroll
    for (int i = 0; i < 8; ++i) {
      const int m = m0 + i + 8 * kh;          // token row (0..16383)
      const int n = n0 + t * 16 + ml;         // feature col (0..3071)
      const int s  = n >> 10;                 // 0:q 1:k 2:v
      const int r  = n & 1023;
      const int h  = r >> 6;
      const int dd = r & 63;
      const int b  = m >> 12;                 // /T_
      const int tt = m & 4095;
      const size_t idx = (((size_t)b * H_ + h) * T_ + tt) * DH_ + dd;
      const float val = acc[t][i];
      if (s == 0)      q_ws[idx] = val * scale;
      else if (s == 1) k_ws[idx] = (val >= 0.0f) ? 1.0f : -1.0f;
      else             v_ws[idx] = (val >= 0.0f) ? 1.0f : -1.0f;
    }
  }
}

// ---------------------------------------------------------------------------
// Kernel 2: chunked attention.  One 8-wave block per (b,h).  Sequential loop
// over 64 chunks carrying prefix state S (64x64 f32) in LDS.  All chunk
// operands (qc,kc,vc) staged in LDS via coalesced float4 loads.  96KB LDS.
// ---------------------------------------------------------------------------
__global__ __launch_bounds__(256)
void attn_kernel(const float* __restrict__ q_ws,
                 const float* __restrict__ k_ws,
                 const float* __restrict__ v_ws,
                 float* __restrict__ out_ws,
                 float* __restrict__ fmat,
                 float* __restrict__ fcnt) {
  __shared__ float Sm[DH_ * DH_];   // prefix sum of chunk_kv
  __shared__ float sc[C_  * C_ ];   // masked scores
  __shared__ float ck[DH_ * DH_];   // current chunk_kv
  __shared__ float qs[C_ * DH_];    // qc tile
  __shared__ float ks[C_ * DH_];    // kc tile
  __shared__ float vs[C_ * DH_];    // vc tile

  const int bh = blockIdx.x;        // 0..63
  const int bb = bh / H_;
  const int h  = bh % H_;
  const float* qb = q_ws + (size_t)bh * T_ * DH_;
  const float* kb = k_ws + (size_t)bh * T_ * DH_;
  const float* vb = v_ws + (size_t)bh * T_ * DH_;

  const int tid  = threadIdx.x;
  const int lane = tid & 31;
  const int wave = tid >> 5;
  const int ml   = lane & 15;
  const int kh   = lane >> 4;

  for (int i = tid; i < DH_ * DH_; i += 256) Sm[i] = 0.0f;
  __syncthreads();

  for (int n = 0; n < NC_; ++n) {
    const float* qn = qb + (size_t)n * C_ * DH_;
    const float* kn = kb + (size_t)n * C_ * DH_;
    const float* vn = vb + (size_t)n * C_ * DH_;

    // ---- phase 0: stage qc,kc,vc into LDS (coalesced float4) ----
    for (int i = tid * 4; i < C_ * DH_; i += 256 * 4) {
      *(float4*)&qs[i] = *(const float4*)&qn[i];
      *(float4*)&ks[i] = *(const float4*)&kn[i];
      *(float4*)&vs[i] = *(const float4*)&vn[i];
    }
    __syncthreads();

    // ---- phase 1: scores + chunk_kv tiles (2 tiles per wave) ----
#pragma unroll
    for (int tt2 = 0; tt2 < 2; ++tt2) {
      const int tile = wave * 2 + tt2;   // 0..15 over 4x4 grid
      const int ti = tile >> 2;
      const int tj = tile & 3;

      // scores tile (causal: tiles above diagonal are zero)
      v8f acc = {};
      if (tj <= ti) {
        const float* arow = &qs[(16 * ti + ml) * DH_];
        const float* brow = &ks[(16 * tj + ml) * DH_];
#pragma unroll 4
        for (int kk = 0; kk < DH_; kk += 4) {
          const int ko = kk + 2 * kh;
          acc = wmma4(*(const v2f*)(arow + ko), *(const v2f*)(brow + ko), acc);
        }
      }
#pragma unroll
      for (int i = 0; i < 8; ++i) {
        const int mi = 16 * ti + i + 8 * kh;
        const int nj = 16 * tj + ml;
        float s = acc[i];
        if (nj > mi) s = 0.0f;            // tril mask (diag kept)
        sc[mi * C_ + nj] = s;
      }

      // chunk_kv tile: ck[kk][dd] = sum_c kc[c][kk] * vc[c][dd]
      v8f acc2 = {};
#pragma unroll 4
      for (int cb = 0; cb < C_; cb += 4) {
        const int co = cb + 2 * kh;
        v2f a, b2;
        a.x  = ks[(co + 0) * DH_ + 16 * ti + ml];
        a.y  = ks[(co + 1) * DH_ + 16 * ti + ml];
        b2.x = vs[(co + 0) * DH_ + 16 * tj + ml];
        b2.y = vs[(co + 1) * DH_ + 16 * tj + ml];
        acc2 = wmma4(a, b2, acc2);
      }
#pragma unroll
      for (int i = 0; i < 8; ++i)
        ck[(16 * ti + i + 8 * kh) * DH_ + 16 * tj + ml] = acc2[i];
    }
    __syncthreads();

    // ---- phase 2: out = (scores@vc + qc@S) / total ----
#pragma unroll
    for (int tt2 = 0; tt2 < 2; ++tt2) {
      const int tile = wave * 2 + tt2;
      const int ti = tile >> 2;
      const int tj = tile & 3;

      v8f acc = {};
      const float* srow = &sc[(16 * ti + ml) * C_];
      const float* qrow = &qs[(16 * ti + ml) * DH_];
#pragma unroll 4
      for (int kk = 0; kk < C_; kk += 4) {
        const int ko = kk + 2 * kh;
        // intra: A = scores rows, B = vc
        v2f a1 = *(const v2f*)(srow + ko);
        v2f b1;
        b1.x = vs[(ko + 0) * DH_ + 16 * tj + ml];
        b1.y = vs[(ko + 1) * DH_ + 16 * tj + ml];
        acc = wmma4(a1, b1, acc);
        // cross: A = qc, B = S (exclusive prefix)
        v2f a2 = *(const v2f*)(qrow + ko);
        v2f b2;
        b2.x = Sm[(ko + 0) * DH_ + 16 * tj + ml];
        b2.y = Sm[(ko + 1) * DH_ + 16 * tj + ml];
        acc = wmma4(a2, b2, acc);
      }
#pragma unroll
      for (int i = 0; i < 8; ++i) {
        const int ig  = 16 * ti + i + 8 * kh;        // row within chunk
        const int tok = n * C_ + ig;
        const float total = (float)(tok + 1);        // >= 1, clip is no-op
        out_ws[((size_t)bb * T_ + tok) * D_ + h * DH_ + 16 * tj + ml] =
            acc[i] / total;
      }
    }
    __syncthreads();

    // ---- phase 3: S += chunk_kv (after all cross reads) ----
    for (int i = tid; i < DH_ * DH_; i += 256) Sm[i] += ck[i];
    __syncthreads();
  }

  // final_matrix = S (inclusive sum over all chunks), final_count = T
  for (int i = tid; i < DH_ * DH_; i += 256)
    fmat[(size_t)bh * DH_ * DH_ + i] = Sm[i];
  if (tid == 0) fcnt[bh] = (float)T_;
}

// ---------------------------------------------------------------------------
// Kernel 3: y = out @ W_o^T  (pipelined WMMA strip-GEMM, plain store)
// ---------------------------------------------------------------------------
__global__ __launch_bounds__(128)
void out_gemm_kernel(const float* __restrict__ outw,
                     const float* __restrict__ Wo,
                     float* __restrict__ y) {
  const int lane = threadIdx.x & 31;
  const int wave = threadIdx.x >> 5;
  const int ml   = lane & 15;
  const int kh   = lane >> 4;

  const int m0 = (blockIdx.y * 4 + wave) * 16;
  const int n0 = blockIdx.x * 64;

  v8f acc[4] = {{}, {}, {}, {}};
  strip_gemm(outw + (size_t)(m0 + ml) * D_,
             Wo + (size_t)(n0 +  0 + ml) * D_,
             Wo + (size_t)(n0 + 16 + ml) * D_,
             Wo + (size_t)(n0 + 32 + ml) * D_,
             Wo + (size_t)(n0 + 48 + ml) * D_,
             2 * kh, acc);

#pragma unroll
  for (int t = 0; t < 4; ++t)
#pragma unroll
    for (int i = 0; i < 8; ++i) {
      const int m = m0 + i + 8 * kh;
      const int n = n0 + t * 16 + ml;
      y[(size_t)m * D_ + n] = acc[t][i];
    }
}

// ---------------------------------------------------------------------------
extern "C" void kernel_launch(void* const* d_in, const int* in_sizes, int n_in,
                              void* d_out, int out_size, void* d_ws, size_t ws_size,
                              hipStream_t stream) {
  const float* x    = (const float*)d_in[0];
  const float* Wqkv = (const float*)d_in[1];
  const float* Wo   = (const float*)d_in[2];

  float* y    = (float*)d_out;                        // [B,T,D]
  float* fmat = y + (size_t)B_ * T_ * D_;             // [B,H,64,64]
  float* fcnt = fmat + (size_t)B_ * H_ * DH_ * DH_;   // [B,H,1,1]

  float* ws = (float*)d_ws;
  const size_t sz = (size_t)B_ * H_ * T_ * DH_;       // 16,777,216 floats
  float* q_ws   = ws;
  float* k_ws   = ws + sz;
  float* v_ws   = ws + 2 * sz;
  float* out_ws = ws + 3 * sz;

  dim3 g1(3 * D_ / 64, (B_ * T_) / 64);               // (48, 256)
  qkv_gemm_kernel<<<g1, 128, 0, stream>>>(x, Wqkv, q_ws, k_ws, v_ws);

  attn_kernel<<<B_ * H_, 256, 0, stream>>>(q_ws, k_ws, v_ws, out_ws, fmat, fcnt);

  dim3 g2(D_ / 64, (B_ * T_) / 64);                   // (16, 256)
  out_gemm_kernel<<<g2, 128, 0, stream>>>(out_ws, Wo, y);
}